// Encoderfix_51634096832564
// MI455X (gfx1250) — compile-verified
//
#include <hip/hip_runtime.h>
#include <math.h>

// ---------------------------------------------------------------------------
// YOLO target encoder for MI455X (gfx1250).
// Memory-bound: ~132 MB of output stores vs ~0.5 MB of reads -> pure store
// bandwidth problem (floor ~5.7us @ 23.3 TB/s). Fill uses the CDNA5 async
// LDS->global store path (b128, ASYNCcnt) when the builtin is available.
// ---------------------------------------------------------------------------

typedef __attribute__((ext_vector_type(4))) int v4i;

namespace {
constexpr int  B_ = 64;
constexpr int  O_ = 100;
constexpr int  A_ = 9;                       // 3 layers x 3 anchors
__device__ constexpr int  FW_[3]    = {128, 64, 32};
__device__ constexpr long LBASE_[3] = {0L,
                                       128L * 128 * 3,
                                       128L * 128 * 3 + 64L * 64 * 3};
constexpr long N_   = 128L * 128 * 3 + 64L * 64 * 3 + 32L * 32 * 3;  // 64512
constexpr long SZ2  = (long)B_ * N_ * 2;     // xcyc / wh / wt section sizes
constexpr long SZ1  = (long)B_ * N_;         // obj / cls section sizes
constexpr long OFF_XCYC = 0;
constexpr long OFF_WH   = SZ2;
constexpr long OFF_OBJ  = 2 * SZ2;
constexpr long OFF_CLS  = 2 * SZ2 + SZ1;
constexpr long OFF_WT   = 2 * SZ2 + 2 * SZ1;
constexpr long TOT      = 3 * SZ2 + 2 * SZ1; // 33,030,144 floats
}

// ---------------------------------------------------------------------------
// 1) Zero-fill the whole output with 128-bit stores. On gfx1250, stage a
//    16B zero chunk per lane in LDS and stream async b128 stores from LDS
//    (fire-and-forget, tracked by ASYNCcnt) - the CDNA5 async data path.
// ---------------------------------------------------------------------------
__global__ void k_zero_fill(float4* __restrict__ out, long n4) {
  __shared__ float4 zbuf[256];
  float4 z;
  z.x = 0.f; z.y = 0.f; z.z = 0.f; z.w = 0.f;
  zbuf[threadIdx.x] = z;
  __syncthreads();

  long i      = (long)blockIdx.x * blockDim.x + threadIdx.x;
  long stride = (long)gridDim.x * blockDim.x;

#if __has_builtin(__builtin_amdgcn_global_store_async_from_lds_b128)
  auto lsrc = (__attribute__((address_space(3))) v4i*)(&zbuf[threadIdx.x]);
  for (; i < n4; i += stride) {
    auto g = (__attribute__((address_space(1))) v4i*)(out + i);
    __builtin_amdgcn_global_store_async_from_lds_b128(g, lsrc, 0, 0);
  }
  asm volatile("s_wait_asynccnt 0" ::: "memory");
#else
  for (; i < n4; i += stride) out[i] = z;
#endif
}

// ---------------------------------------------------------------------------
// 2) Ignore scatter: obj = -1 where iou >= 0.5 and box valid.
//    One thread per (b, a, o). Bounds checks emulate JAX mode='drop'.
// ---------------------------------------------------------------------------
__global__ void k_ignore(const float* __restrict__ ious,
                         const float* __restrict__ boxes,
                         float* __restrict__ obj,
                         const int* __restrict__ p_inh,
                         const int* __restrict__ p_inw) {
  int t = blockIdx.x * blockDim.x + threadIdx.x;
  if (t >= B_ * A_ * O_) return;
  int o = t % O_;
  int a = (t / O_) % A_;
  int b = t / (O_ * A_);

  float iou = ious[((long)b * A_ + a) * O_ + o];
  if (iou < 0.5f) return;

  const float* bx = boxes + ((long)b * O_ + o) * 4;
  float xmin = bx[0], ymin = bx[1], xmax = bx[2], ymax = bx[3];
  float w = xmax - xmin, h = ymax - ymin;
  // NOTE: reference computes gtx = (xmin + w)/2 (i.e. xmax/2) - replicate.
  float gtx = (xmin + w) * 0.5f;
  float gty = (ymin + h) * 0.5f;
  if (gtx == -1.0f && gty == -1.0f && w == 0.0f && h == 0.0f) return; // invalid

  float inw = (float)p_inw[0];
  float inh = (float)p_inh[0];
  int layer = a / 3;
  int al    = a - layer * 3;
  int fw    = FW_[layer];

  float fx = gtx / inw * (float)fw;
  float fy = gty / inh * (float)fw;
  int lx = (int)fx, ly = (int)fy;
  if (lx < 0 || lx >= fw || ly < 0 || ly >= fw) return;  // mode='drop'

  long cell = (long)ly * fw + lx;
  obj[(long)b * N_ + LBASE_[layer] + cell * 3 + al] = -1.0f;
}

// ---------------------------------------------------------------------------
// 3) Match scatter: one thread per (b, o). Writes xcyc, wh, wt, obj=1, cls.
//    Runs after k_ignore so obj=1.0 overrides -1.0 (as in the reference).
// ---------------------------------------------------------------------------
__global__ void k_match(const int* __restrict__ matches,
                        const float* __restrict__ boxes,
                        const int* __restrict__ gt_ids,
                        const float* __restrict__ anc0,
                        const float* __restrict__ anc1,
                        const float* __restrict__ anc2,
                        float* __restrict__ out,
                        const int* __restrict__ p_inh,
                        const int* __restrict__ p_inw) {
  int t = blockIdx.x * blockDim.x + threadIdx.x;
  if (t >= B_ * O_) return;
  int o = t % O_;
  int b = t / O_;

  const float* bx = boxes + ((long)b * O_ + o) * 4;
  float xmin = bx[0], ymin = bx[1], xmax = bx[2], ymax = bx[3];
  float w = xmax - xmin, h = ymax - ymin;
  float gtx = (xmin + w) * 0.5f;
  float gty = (ymin + h) * 0.5f;
  if (gtx == -1.0f && gty == -1.0f && w == 0.0f && h == 0.0f) return; // invalid

  int m = matches[(long)b * O_ + o];
  if (m < 0 || m >= A_) return;
  int layer = m / 3;
  int al    = m - layer * 3;
  const float* anc = (layer == 0) ? anc0 : (layer == 1) ? anc1 : anc2;
  float aw = anc[al * 2 + 0];
  float ah = anc[al * 2 + 1];

  float inw = (float)p_inw[0];
  float inh = (float)p_inh[0];
  int fw = FW_[layer];

  float fx = gtx / inw * (float)fw;
  float fy = gty / inh * (float)fw;
  int lx = (int)fx, ly = (int)fy;
  if (lx < 0 || lx >= fw || ly < 0 || ly >= fw) return;  // mode='drop'

  long cell = (long)ly * fw + lx;
  long n  = LBASE_[layer] + cell * 3 + al;
  long i1 = (long)b * N_ + n;
  long i2 = i1 * 2;

  out[OFF_XCYC + i2]     = fx - (float)lx;
  out[OFF_XCYC + i2 + 1] = fy - (float)ly;
  out[OFF_WH + i2]       = logf(fmaxf(w, 1.0f) / aw);
  out[OFF_WH + i2 + 1]   = logf(fmaxf(h, 1.0f) / ah);
  float wt = 2.0f - w * h / inw / inh;
  out[OFF_WT + i2]       = wt;
  out[OFF_WT + i2 + 1]   = wt;
  out[OFF_OBJ + i1]      = 1.0f;
  out[OFF_CLS + i1]      = (float)gt_ids[(long)b * O_ + o];
}

extern "C" void kernel_launch(void* const* d_in, const int* in_sizes, int n_in,
                              void* d_out, int out_size, void* d_ws, size_t ws_size,
                              hipStream_t stream) {
  (void)in_sizes; (void)n_in; (void)out_size; (void)d_ws; (void)ws_size;

  const int*   matches  = (const int*)d_in[0];
  const float* ious     = (const float*)d_in[1];
  // d_in[2..4] = out0/out1/out2 (shape carriers only, contents unused)
  const float* anc0     = (const float*)d_in[5];
  const float* anc1     = (const float*)d_in[6];
  const float* anc2     = (const float*)d_in[7];
  const float* gt_boxes = (const float*)d_in[8];
  const int*   gt_ids   = (const int*)d_in[9];
  const int*   p_inh    = (const int*)d_in[10];
  const int*   p_inw    = (const int*)d_in[11];

  float* out = (float*)d_out;

  // 1) zero fill (132 MB of b128 stores; bandwidth-bound)
  long n4 = TOT / 4;
  k_zero_fill<<<8192, 256, 0, stream>>>((float4*)out, n4);

  // 2) ignore scatter
  int tIg = B_ * A_ * O_;
  k_ignore<<<(tIg + 255) / 256, 256, 0, stream>>>(
      ious, gt_boxes, out + OFF_OBJ, p_inh, p_inw);

  // 3) match scatter (overrides obj)
  int tM = B_ * O_;
  k_match<<<(tM + 255) / 256, 256, 0, stream>>>(
      matches, gt_boxes, gt_ids, anc0, anc1, anc2, out, p_inh, p_inw);
}